// Model_10402410791269
// MI455X (gfx1250) — compile-verified
//
#include <hip/hip_runtime.h>
#include <math.h>
#include <stdint.h>

// ---------------------------------------------------------------------------
// 3-layer LSTM LM for MI455X (gfx1250, wave32, WMMA f32).
//  - embed gather
//  - per layer: Xproj GEMM (parallel WMMA, 2x2 reg tiling, async-LDS staging)
//    then resident-workgroup recurrence with Whh pinned in 320KB WGP LDS
//  - FC GEMM [8192 x 32000], K=128: store-bound (1.05GB / 23.3TB/s), fp32
//    WMMA suffices; logits stored non-temporally (bigger than L2, write-once).
// ---------------------------------------------------------------------------

typedef float v2f __attribute__((ext_vector_type(2)));
typedef float v4f __attribute__((ext_vector_type(4)));
typedef float v8f __attribute__((ext_vector_type(8)));

#define SEQ    256
#define BATCH  32
#define VOCAB  32000
#define H      128
#define NGATE  512        // 4*H
#define LAYERS 3
#define LDS_PAD 132       // row stride (floats): conflict-free strided frag loads

__device__ __forceinline__ float sigmoidf_(float x) {
    return 1.0f / (1.0f + __expf(-x));
}

// Async DMA: global (per-lane 16B) -> LDS, tracked by ASYNCcnt.
// lds_off = wave-relative LDS byte offset (low 32 bits of generic shared ptr).
__device__ __forceinline__ void async_b128(void* lds_ptr, const float* gaddr) {
    uint32_t lds_off = (uint32_t)(uintptr_t)lds_ptr;
    asm volatile("global_load_async_to_lds_b128 %0, %1, off"
                 :: "v"(lds_off), "v"(gaddr) : "memory");
}
__device__ __forceinline__ void wait_async0() {
    asm volatile("s_wait_asynccnt 0x0" ::: "memory");
}

// ---------------------------------------------------------------------------
// Embedding gather: out[row][:] = emb[x[row]][:]
// ---------------------------------------------------------------------------
__global__ void embed_kernel(const int* __restrict__ x,
                             const float* __restrict__ emb,
                             float* __restrict__ out) {
    const int wave = threadIdx.x >> 5;
    const int lane = threadIdx.x & 31;
    const int row  = blockIdx.x * 8 + wave;
    const int tok  = x[row];
    v4f v = *(const v4f*)(emb + (size_t)tok * H + lane * 4);
    *(v4f*)(out + (size_t)row * H + lane * 4) = v;
}

// ---------------------------------------------------------------------------
// C[M,N] = A[M,128] @ W[N,128]^T + b1[N] (+ b2[N])
// grid (N/128, M/128), block 512 = 16 waves (4x4 wave grid).
// Each wave: 2x2 tiles of 16x16 (32x32 macro-tile) -> 4 independent WMMA
// accumulator chains, 2 WMMAs per fragment load. Blocks staged with async DMA.
// ---------------------------------------------------------------------------
template <bool NT>
__global__ void gemm_k128(const float* __restrict__ A,
                          const float* __restrict__ W,
                          const float* __restrict__ b1,
                          const float* __restrict__ b2,
                          float* __restrict__ C, int N) {
    extern __shared__ float smem[];
    float* As = smem;                   // [128][LDS_PAD]
    float* Ws = smem + 128 * LDS_PAD;   // [128][LDS_PAD]

    const int tid  = threadIdx.x;
    const int mblk = blockIdx.y * 128;
    const int nblk = blockIdx.x * 128;

    // Stage 128x128 of A and W via async DMA: 4096 float4 each, 8/thread.
    #pragma unroll
    for (int i = 0; i < 8; ++i) {
        int f  = tid + i * 512;
        int r  = f >> 5;              // 32 float4 per row
        int c4 = f & 31;
        async_b128(As + r * LDS_PAD + c4 * 4, A + (size_t)(mblk + r) * H + c4 * 4);
        async_b128(Ws + r * LDS_PAD + c4 * 4, W + (size_t)(nblk + r) * H + c4 * 4);
    }
    wait_async0();
    __syncthreads();

    const int wave = tid >> 5;
    const int lane = tid & 31;
    const int lo   = lane & 15;
    const int hi   = lane >> 4;
    const int m0   = (wave >> 2) * 32;    // 32-row macro-tile
    const int n0   = (wave & 3) * 32;     // 32-col macro-tile

    // Fragment base pointers (A rows m, W rows n; K pair selected by hi).
    const float* ap0 = As + (m0 + lo) * LDS_PAD + 2 * hi;
    const float* ap1 = As + (m0 + 16 + lo) * LDS_PAD + 2 * hi;
    const float* wp0 = Ws + (n0 + lo) * LDS_PAD + 2 * hi;
    const float* wp1 = Ws + (n0 + 16 + lo) * LDS_PAD + 2 * hi;

    v8f acc00 = {}, acc01 = {}, acc10 = {}, acc11 = {};
    #pragma unroll
    for (int kk = 0; kk < 32; ++kk) {
        v2f a0; a0.x = ap0[kk * 4]; a0.y = ap0[kk * 4 + 1];
        v2f a1; a1.x = ap1[kk * 4]; a1.y = ap1[kk * 4 + 1];
        v2f b0; b0.x = wp0[kk * 4]; b0.y = wp0[kk * 4 + 1];
        v2f b1v; b1v.x = wp1[kk * 4]; b1v.y = wp1[kk * 4 + 1];
        acc00 = __builtin_amdgcn_wmma_f32_16x16x4_f32(false, a0, false, b0,  (short)0, acc00, false, false);
        acc01 = __builtin_amdgcn_wmma_f32_16x16x4_f32(false, a0, false, b1v, (short)0, acc01, false, false);
        acc10 = __builtin_amdgcn_wmma_f32_16x16x4_f32(false, a1, false, b0,  (short)0, acc10, false, false);
        acc11 = __builtin_amdgcn_wmma_f32_16x16x4_f32(false, a1, false, b1v, (short)0, acc11, false, false);
    }

    const int nA = nblk + n0 + lo;
    const int nB = nA + 16;
    float biasA = b1 ? b1[nA] : 0.0f;
    float biasB = b1 ? b1[nB] : 0.0f;
    if (b2) { biasA += b2[nA]; biasB += b2[nB]; }

    float* cp0 = C + (size_t)(mblk + m0 + 8 * hi) * N;        // rows for tiles *0*
    float* cp1 = C + (size_t)(mblk + m0 + 16 + 8 * hi) * N;   // rows for tiles *1*
    #pragma unroll
    for (int r = 0; r < 8; ++r) {
        float v00 = acc00[r] + biasA;
        float v01 = acc01[r] + biasB;
        float v10 = acc10[r] + biasA;
        float v11 = acc11[r] + biasB;
        if (NT) {
            __builtin_nontemporal_store(v00, cp0 + (size_t)r * N + nA);
            __builtin_nontemporal_store(v01, cp0 + (size_t)r * N + nB);
            __builtin_nontemporal_store(v10, cp1 + (size_t)r * N + nA);
            __builtin_nontemporal_store(v11, cp1 + (size_t)r * N + nB);
        } else {
            cp0[(size_t)r * N + nA] = v00;
            cp0[(size_t)r * N + nB] = v01;
            cp1[(size_t)r * N + nA] = v10;
            cp1[(size_t)r * N + nB] = v11;
        }
    }
}

// ---------------------------------------------------------------------------
// Recurrence for one layer. Single workgroup, 512 threads = 16 waves.
// Wave (mt = w>>3, jt = w&7): batch rows mt*16..+15, hidden cols jt*16..+15.
// Each wave accumulates ALL FOUR gate quadrants (4 independent WMMA chains)
// so the cell update is wave-local. Whh (512x132 fp32 ~270KB) pinned in the
// CDNA5 WGP's 320KB LDS for the whole kernel.
// ---------------------------------------------------------------------------
__global__ void lstm_kernel(const float* __restrict__ G,    // [SEQ*BATCH, 512] x-proj + biases
                            const float* __restrict__ Whh,  // [512,128]
                            const float* __restrict__ h0,   // [32,128]
                            const float* __restrict__ c0,   // [32,128]
                            float* __restrict__ Xout,       // [SEQ*BATCH, 128]
                            float* __restrict__ hT,         // [32,128]
                            float* __restrict__ cT) {       // [32,128]
    extern __shared__ float smem[];
    float* Wl = smem;                    // [512][LDS_PAD]
    float* hl = smem + NGATE * LDS_PAD;  // [32][LDS_PAD]
    const int tid = threadIdx.x;

    // Stage Whh (16384 float4) and h0 (1024 float4) via async DMA.
    #pragma unroll
    for (int i = 0; i < 32; ++i) {
        int f  = tid + i * 512;
        int r  = f >> 5;
        int c4 = f & 31;
        async_b128(Wl + r * LDS_PAD + c4 * 4, Whh + (size_t)r * H + c4 * 4);
    }
    #pragma unroll
    for (int i = 0; i < 2; ++i) {
        int f  = tid + i * 512;
        int r  = f >> 5;
        int c4 = f & 31;
        async_b128(hl + r * LDS_PAD + c4 * 4, h0 + (size_t)r * H + c4 * 4);
    }

    const int wave = tid >> 5;
    const int lane = tid & 31;
    const int lo   = lane & 15;
    const int hi   = lane >> 4;
    const int m0   = (wave >> 3) * 16;   // batch tile base (0 or 16)
    const int j0   = (wave & 7) * 16;    // hidden tile base (0..112)

    // Cell state: this lane's 8 elements of the 16x16 (m,j) tile (C/D layout).
    float cst[8];
    #pragma unroll
    for (int r = 0; r < 8; ++r)
        cst[r] = c0[(m0 + r + 8 * hi) * H + j0 + lo];

    const float* ap  = hl + (m0 + lo) * LDS_PAD + 2 * hi;
    const float* wpI = Wl + (0 * H + j0 + lo) * LDS_PAD + 2 * hi;
    const float* wpF = Wl + (1 * H + j0 + lo) * LDS_PAD + 2 * hi;
    const float* wpG = Wl + (2 * H + j0 + lo) * LDS_PAD + 2 * hi;
    const float* wpO = Wl + (3 * H + j0 + lo) * LDS_PAD + 2 * hi;

    wait_async0();
    __syncthreads();

    for (int t = 0; t < SEQ; ++t) {
        // Init accumulators from precomputed x-projection (biases included).
        const float* g = G + (size_t)t * BATCH * NGATE;
        v8f ai = {}, af = {}, ag = {}, ao = {};
        #pragma unroll
        for (int r = 0; r < 8; ++r) {
            const int m = m0 + r + 8 * hi;
            const float* gr = g + (size_t)m * NGATE + j0 + lo;
            ai[r] = gr[0];
            af[r] = gr[H];
            ag[r] = gr[2 * H];
            ao[r] = gr[3 * H];
        }
        // Prefetch next step's x-projection rows (latency-critical loop).
        if (t + 1 < SEQ) {
            const float* gn = g + (size_t)BATCH * NGATE + (size_t)(m0 + 8 * hi) * NGATE + j0 + lo;
            __builtin_prefetch(gn, 0, 0);
            __builtin_prefetch(gn + H, 0, 0);
            __builtin_prefetch(gn + 2 * H, 0, 0);
            __builtin_prefetch(gn + 3 * H, 0, 0);
        }
        // gates += h_prev @ Whh^T  (four independent WMMA chains -> ILP)
        #pragma unroll
        for (int kk = 0; kk < 32; ++kk) {
            v2f a; a.x = ap[kk * 4]; a.y = ap[kk * 4 + 1];
            v2f bi; bi.x = wpI[kk * 4]; bi.y = wpI[kk * 4 + 1];
            ai = __builtin_amdgcn_wmma_f32_16x16x4_f32(false, a, false, bi, (short)0, ai, false, false);
            v2f bf; bf.x = wpF[kk * 4]; bf.y = wpF[kk * 4 + 1];
            af = __builtin_amdgcn_wmma_f32_16x16x4_f32(false, a, false, bf, (short)0, af, false, false);
            v2f bg; bg.x = wpG[kk * 4]; bg.y = wpG[kk * 4 + 1];
            ag = __builtin_amdgcn_wmma_f32_16x16x4_f32(false, a, false, bg, (short)0, ag, false, false);
            v2f bo; bo.x = wpO[kk * 4]; bo.y = wpO[kk * 4 + 1];
            ao = __builtin_amdgcn_wmma_f32_16x16x4_f32(false, a, false, bo, (short)0, ao, false, false);
        }
        __syncthreads();   // everyone done reading h_prev from LDS

        float hv[8];
        #pragma unroll
        for (int r = 0; r < 8; ++r) {
            float gi = sigmoidf_(ai[r]);
            float gf = sigmoidf_(af[r]);
            float gg = tanhf(ag[r]);
            float go = sigmoidf_(ao[r]);
            cst[r] = gf * cst[r] + gi * gg;
            hv[r]  = go * tanhf(cst[r]);
        }
        #pragma unroll
        for (int r = 0; r < 8; ++r) {
            const int m = m0 + r + 8 * hi;
            hl[m * LDS_PAD + j0 + lo] = hv[r];
            Xout[((size_t)t * BATCH + m) * H + j0 + lo] = hv[r];
        }
        __syncthreads();   // new h visible for next step
    }

    #pragma unroll
    for (int r = 0; r < 8; ++r) {
        const int m = m0 + r + 8 * hi;
        hT[m * H + j0 + lo] = hl[m * LDS_PAD + j0 + lo];
        cT[m * H + j0 + lo] = cst[r];
    }
}

// ---------------------------------------------------------------------------
extern "C" void kernel_launch(void* const* d_in, const int* in_sizes, int n_in,
                              void* d_out, int out_size, void* d_ws, size_t ws_size,
                              hipStream_t stream) {
    const int*   x    = (const int*)  d_in[0];
    const float* h0   = (const float*)d_in[1];
    const float* c0   = (const float*)d_in[2];
    const float* emb  = (const float*)d_in[3];
    const float* Wih  = (const float*)d_in[4];
    const float* Whh  = (const float*)d_in[5];
    const float* bih  = (const float*)d_in[6];
    const float* bhh  = (const float*)d_in[7];
    const float* fc_w = (const float*)d_in[8];
    const float* fc_b = (const float*)d_in[9];
    float* out = (float*)d_out;

    const size_t SB = (size_t)SEQ * BATCH;          // 8192
    float* Xa = (float*)d_ws;                       // [8192,128]  4 MB
    float* Xb = Xa + SB * H;                        // [8192,128]  4 MB
    float* G  = Xb + SB * H;                        // [8192,512] 16 MB

    embed_kernel<<<dim3(SB / 8), dim3(256), 0, stream>>>(x, emb, Xa);

    const size_t gemm_smem = (size_t)2 * 128 * LDS_PAD * sizeof(float);          // ~135 KB
    const size_t lstm_smem = (size_t)(NGATE + BATCH) * LDS_PAD * sizeof(float);  // ~280 KB
    const size_t LOGITS = SB * (size_t)VOCAB;

    float* Xcur = Xa;
    float* Xnext = Xb;
    for (int l = 0; l < LAYERS; ++l) {
        dim3 gg(NGATE / 128, SB / 128);   // (4, 64)
        gemm_k128<false><<<gg, dim3(512), gemm_smem, stream>>>(
            Xcur, Wih + (size_t)l * NGATE * H,
            bih + (size_t)l * NGATE, bhh + (size_t)l * NGATE, G, NGATE);

        lstm_kernel<<<dim3(1), dim3(512), lstm_smem, stream>>>(
            G, Whh + (size_t)l * NGATE * H,
            h0 + (size_t)l * BATCH * H, c0 + (size_t)l * BATCH * H,
            Xnext,
            out + LOGITS + (size_t)l * BATCH * H,
            out + LOGITS + (size_t)LAYERS * BATCH * H + (size_t)l * BATCH * H);

        float* tmp = Xcur; Xcur = Xnext; Xnext = tmp;
    }

    dim3 gf(VOCAB / 128, SB / 128);       // (250, 64)
    gemm_k128<true><<<gf, dim3(512), gemm_smem, stream>>>(Xcur, fc_w, fc_b, nullptr, out, VOCAB);
}